// WeightedBCELoss_70128226009669
// MI455X (gfx1250) — compile-verified
//
#include <hip/hip_runtime.h>

// Weighted BCE, elementwise over [B,S] fp32:
//   label==0 -> -log1p(-p)        == -log(1-p)   (p in (0.01,0.99), safe)
//   label==1 -> -w[row] * log(p)
//
// Memory-bound: 402 MB moved per call -> ~17us at 23.3 TB/s. Strategy:
// b128 NT loads/stores, one log per element, scalar weight load per row.

// Native clang vectors: required by __builtin_nontemporal_load/store and
// lower to single b128 vector-memory ops.
typedef float v4f __attribute__((ext_vector_type(4)));
typedef int   v4i __attribute__((ext_vector_type(4)));

__device__ __forceinline__ float bce1(float p, int lab, float w) {
    // Fold the two branches into a single transcendental:
    //   arg   = lab ? p  : (1 - p)
    //   scale = lab ? -w : -1
    //   out   = scale * log(arg)
    const bool z = (lab == 0);
    const float arg   = z ? (1.0f - p) : p;
    const float scale = z ? -1.0f      : -w;
    return scale * __logf(arg);   // v_log_f32 (+ ln2 mul), TRANS co-exec
}

__device__ __forceinline__ v4f bce4(v4f p, v4i l, float w) {
    v4f r;
    r.x = bce1(p.x, l.x, w);
    r.y = bce1(p.y, l.y, w);
    r.z = bce1(p.z, l.z, w);
    r.w = bce1(p.w, l.w, w);
    return r;
}

// Fast path: one block per row, ITERS float4 per thread (ITERS*1024 floats/row).
template <int ITERS>
__global__ __launch_bounds__(256) void bce_row_kernel(
    const float* __restrict__ pred,
    const int*   __restrict__ labels,
    const float* __restrict__ weight,
    float*       __restrict__ out,
    int cols)
{
    const int row = blockIdx.x;
    const float w = weight[row];                 // wave-uniform -> s_load_b32

    const size_t base = (size_t)row * (size_t)cols;
    const v4f* __restrict__ p4 = reinterpret_cast<const v4f*>(pred + base);
    const v4i* __restrict__ l4 = reinterpret_cast<const v4i*>(labels + base);
    v4f*       __restrict__ o4 = reinterpret_cast<v4f*>(out + base);

    const int t = threadIdx.x;

    v4f p[ITERS];
    v4i l[ITERS];

    // Issue all 2*ITERS b128 loads back-to-back (NT: streaming, no reuse,
    // working set > 192MB L2) so one s_wait covers max outstanding loads.
#pragma unroll
    for (int i = 0; i < ITERS; ++i) {
        const int idx = t + i * 256;
        p[i] = __builtin_nontemporal_load(p4 + idx);
        l[i] = __builtin_nontemporal_load(l4 + idx);
    }

#pragma unroll
    for (int i = 0; i < ITERS; ++i) {
        const int idx = t + i * 256;
        const v4f r = bce4(p[i], l[i], w);
        __builtin_nontemporal_store(r, o4 + idx);  // global_store_b128 TH=NT
    }
}

// Generic fallback: grid-stride over float4s, row derived per vector.
__global__ __launch_bounds__(256) void bce_flat_kernel(
    const v4f*  __restrict__ p4,
    const v4i*  __restrict__ l4,
    const float* __restrict__ weight,
    v4f*        __restrict__ o4,
    int nvec, int vec_per_row)
{
    int i = blockIdx.x * blockDim.x + threadIdx.x;
    const int stride = gridDim.x * blockDim.x;
    for (; i < nvec; i += stride) {
        const int row = i / vec_per_row;
        const float w = weight[row];
        const v4f p = __builtin_nontemporal_load(p4 + i);
        const v4i l = __builtin_nontemporal_load(l4 + i);
        const v4f r = bce4(p, l, w);
        __builtin_nontemporal_store(r, o4 + i);
    }
}

extern "C" void kernel_launch(void* const* d_in, const int* in_sizes, int n_in,
                              void* d_out, int out_size, void* d_ws, size_t ws_size,
                              hipStream_t stream) {
    const float* pred   = (const float*)d_in[0];
    const int*   labels = (const int*)d_in[1];
    const float* weight = (const float*)d_in[2];
    float*       out    = (float*)d_out;

    const int total = in_sizes[0];          // B * S
    const int rows  = in_sizes[2];          // B (weight is [B])
    const int cols  = (rows > 0) ? total / rows : 0;   // S

    // Fast path: S = 8192 -> 2048 float4/row -> 8 float4/thread @ 256 threads.
    if (rows > 0 && cols == 8192 && (size_t)rows * cols == (size_t)total) {
        bce_row_kernel<8><<<dim3(rows), dim3(256), 0, stream>>>(
            pred, labels, weight, out, cols);
        return;
    }

    // Generic fallback (cols must be a multiple of 4 for float4 path;
    // reference shapes guarantee this).
    const int nvec = total / 4;
    const int vec_per_row = cols / 4;
    int blocks = (nvec + 255) / 256;
    if (blocks > 65535 * 8) blocks = 65535 * 8;
    bce_flat_kernel<<<dim3(blocks), dim3(256), 0, stream>>>(
        (const v4f*)pred, (const v4i*)labels, weight, (v4f*)out,
        nvec, vec_per_row);
}